// UserItemWithRepostEmbeddingTrimmed_70755291234322
// MI455X (gfx1250) — compile-verified
//
#include <hip/hip_runtime.h>
#include <hip/hip_bf16.h>
#include <stdint.h>

// Problem constants (match reference)
#define NTOKEN 200000
#define NINP   64
#define NHID   128
#define NEDGE  2000000
#define NROWS  (64 * 200)   // B*L

typedef __attribute__((ext_vector_type(2))) float v2f;
typedef __attribute__((ext_vector_type(8))) float v8f;

// ---------------------------------------------------------------------------
// 1) Degree / normalization:  deg[i] = 1 (self loop) + #edges with dst==i,
//    then dinv[i] = rsqrt(deg[i]) in place.
// ---------------------------------------------------------------------------
__global__ void init_deg_kernel(float* __restrict__ deg) {
    int i = blockIdx.x * blockDim.x + threadIdx.x;
    if (i < NTOKEN) deg[i] = 1.0f;
}

__global__ void deg_edges_kernel(const long long* __restrict__ dst, float* __restrict__ deg) {
    int e = blockIdx.x * blockDim.x + threadIdx.x;
    if (e < NEDGE) atomicAdd(&deg[(int)dst[e]], 1.0f);
}

__global__ void finalize_dinv_kernel(float* __restrict__ dinv) {
    int i = blockIdx.x * blockDim.x + threadIdx.x;
    if (i < NTOKEN) dinv[i] = rsqrtf(dinv[i]);
}

// ---------------------------------------------------------------------------
// 2) Dense GEMM via V_WMMA_F32_16X16X4_F32, weights staged through LDS.
//    A: [M,K] row-major, Bw: [K,N] row-major (K*N*4 <= 32KB), C: [M,N].
//    One wave per 16-row strip, NT = N/16 accumulators (all columns), K step 4.
//
//    f32 WMMA VGPR layout (ISA 7.12.2):
//      A 16x4 : lanes 0-15 hold row M=lane, VGPR0/1 = K0/K1; lanes 16-31 = K2/K3
//      B 4x16 : VGPR v, lanes 0-15 = row K=v, lanes 16-31 = row K=v+2
//      C 16x16: VGPR v, lanes 0-15 -> M=v, lanes 16-31 -> M=v+8; col = lane&15
// ---------------------------------------------------------------------------
template <int NT, int K>
__global__ void gemm_wmma_kernel(const float* __restrict__ A, const float* __restrict__ Bw,
                                 float* __restrict__ C, int M) {
    constexpr int N = NT * 16;
    __shared__ float sB[K * N];

    // Cooperative, coalesced staging of the whole weight matrix into LDS.
    for (int i = threadIdx.x; i < K * N; i += blockDim.x) sB[i] = Bw[i];
    __syncthreads();

    const int wave = (blockIdx.x * blockDim.x + threadIdx.x) >> 5;
    const int lane = threadIdx.x & 31;
    const int m0   = wave * 16;
    if (m0 >= M) return;                 // wave-uniform: EXEC stays all-1s for WMMA

    const int row   = lane & 15;
    const int khalf = (lane >> 4) << 1;  // 0 for lanes 0-15, 2 for lanes 16-31

    v8f acc[NT];
    v8f zero = {};
#pragma unroll
    for (int t = 0; t < NT; ++t) acc[t] = zero;

    const float* aRow = A + (size_t)(m0 + row) * K + khalf;

#pragma unroll 4
    for (int k0 = 0; k0 < K; k0 += 4) {
        v2f a;
        a.x = aRow[k0];
        a.y = aRow[k0 + 1];
        const float* bp0 = &sB[(k0 + khalf) * N + row];
#pragma unroll
        for (int t = 0; t < NT; ++t) {
            v2f b;
            b.x = bp0[t * 16];           // row k0+khalf
            b.y = bp0[t * 16 + N];       // row k0+khalf+1
            acc[t] = __builtin_amdgcn_wmma_f32_16x16x4_f32(
                false, a, false, b, (short)0, acc[t], false, false);
        }
    }

    const int srow = (lane >> 4) << 3;   // 0 or 8
#pragma unroll
    for (int t = 0; t < NT; ++t) {
        float* cp = C + (size_t)(m0 + srow) * N + t * 16 + row;
#pragma unroll
        for (int v = 0; v < 8; ++v) cp[(size_t)v * N] = acc[t][v];
    }
}

// ---------------------------------------------------------------------------
// 3) Self-loop init:  out[i,f] = dinv[i]^2 * xw[i,f] + bias[f]
//    (also serves as the zero-init for the atomic scatter that follows)
// ---------------------------------------------------------------------------
template <int F>
__global__ void self_init_kernel(const float* __restrict__ dinv, const float* __restrict__ xw,
                                 const float* __restrict__ bias, float* __restrict__ out) {
    long i = (long)blockIdx.x * blockDim.x + threadIdx.x;
    if (i >= (long)NTOKEN * F) return;
    int node = (int)(i / F);
    int f    = (int)(i % F);
    float di = dinv[node];
    out[i] = di * di * xw[i] + bias[f];
}

// ---------------------------------------------------------------------------
// 4) Edge aggregation: one wave32 per edge.
//    Lane l handles a contiguous float4 (F=128) / float2 (F=64) of features:
//    coalesced 512B/256B gather of xw[src], vector-adjacent f32 atomics to out[dst].
//    xw (102.4 MB) fits in the 192 MB L2, so the random gathers are L2-resident.
// ---------------------------------------------------------------------------
template <int F>
__global__ void edge_agg_kernel(const long long* __restrict__ src, const long long* __restrict__ dst,
                                const float* __restrict__ dinv, const float* __restrict__ xw,
                                float* __restrict__ out) {
    const int wave = (blockIdx.x * blockDim.x + threadIdx.x) >> 5;
    const int lane = threadIdx.x & 31;
    if (wave >= NEDGE) return;
    const int s = (int)src[wave];
    const int d = (int)dst[wave];
    const float w = dinv[s] * dinv[d];
    if (F == 128) {
        float4 v = ((const float4*)(xw + (size_t)s * 128))[lane];
        float* op = out + (size_t)d * 128 + lane * 4;
        atomicAdd(op + 0, v.x * w);
        atomicAdd(op + 1, v.y * w);
        atomicAdd(op + 2, v.z * w);
        atomicAdd(op + 3, v.w * w);
    } else {
        float2 v = ((const float2*)(xw + (size_t)s * 64))[lane];
        float* op = out + (size_t)d * 64 + lane * 2;
        atomicAdd(op + 0, v.x * w);
        atomicAdd(op + 1, v.y * w);
    }
}

// ---------------------------------------------------------------------------
// 5) Final gather: out[r,:] = node_emb[input[r],:]   (r = b*L + l), float2/lane
// ---------------------------------------------------------------------------
__global__ void gather_kernel(const long long* __restrict__ idx, const float* __restrict__ node_emb,
                              float* __restrict__ out) {
    const int t = blockIdx.x * blockDim.x + threadIdx.x;
    const int r = t >> 5;
    const int lane = t & 31;
    if (r >= NROWS) return;
    const long long n = idx[r];
    float2 v = ((const float2*)(node_emb + (size_t)n * NINP))[lane];
    ((float2*)(out + (size_t)r * NINP))[lane] = v;
}

// ---------------------------------------------------------------------------
extern "C" void kernel_launch(void* const* d_in, const int* in_sizes, int n_in,
                              void* d_out, int out_size, void* d_ws, size_t ws_size,
                              hipStream_t stream) {
    // Inputs (setup_inputs order): input, input_timestamp, input_id, edge_index,
    //                              emb, W1, b1, W2, b2
    const long long* input = (const long long*)d_in[0];
    const long long* edge  = (const long long*)d_in[3];   // [2, NEDGE] int64
    const float*     emb   = (const float*)d_in[4];       // [NTOKEN, 64]
    const float*     W1    = (const float*)d_in[5];       // [64, 128]
    const float*     b1    = (const float*)d_in[6];       // [128]
    const float*     W2    = (const float*)d_in[7];       // [128, 64]
    const float*     b2    = (const float*)d_in[8];       // [64]
    float*           out   = (float*)d_out;               // [B, L, 64]

    const long long* esrc = edge;
    const long long* edst = edge + NEDGE;

    // Workspace layout:
    //   dinv : NTOKEN f32                         (0.8 MB)
    //   bufA : NTOKEN*128 f32  (xw, later hw)     (102.4 MB)
    //   bufB : NTOKEN*128 f32  (h, later node_emb)(102.4 MB)
    char*  ws   = (char*)d_ws;
    float* dinv = (float*)ws;
    float* bufA = (float*)(ws + (size_t)NTOKEN * 4);
    float* bufB = (float*)(ws + (size_t)NTOKEN * 4 + (size_t)NTOKEN * NHID * 4);

    const int TB = 256;

    // --- degrees & symmetric normalization -------------------------------
    init_deg_kernel<<<(NTOKEN + TB - 1) / TB, TB, 0, stream>>>(dinv);
    deg_edges_kernel<<<(NEDGE + TB - 1) / TB, TB, 0, stream>>>(edst, dinv);
    finalize_dinv_kernel<<<(NTOKEN + TB - 1) / TB, TB, 0, stream>>>(dinv);

    // --- layer 1: xw = emb @ W1  (WMMA f32, W1 in LDS) -------------------
    {
        int waves  = NTOKEN / 16;                  // 12500 strips
        int blocks = (waves * 32 + TB - 1) / TB;
        gemm_wmma_kernel<8, NINP><<<blocks, TB, 0, stream>>>(emb, W1, bufA, NTOKEN);
    }
    // h = dinv^2 * xw + b1   (self-loop + init)
    self_init_kernel<128><<<(int)(((long)NTOKEN * 128 + TB - 1) / TB), TB, 0, stream>>>(
        dinv, bufA, b1, bufB);
    // h[dst] += dinv[src]*dinv[dst] * xw[src]   (one wave per edge)
    edge_agg_kernel<128><<<(int)(((long)NEDGE * 32 + TB - 1) / TB), TB, 0, stream>>>(
        esrc, edst, dinv, bufA, bufB);

    // --- layer 2: hw = h @ W2  (WMMA f32, W2 in LDS), hw overwrites bufA -
    {
        int waves  = NTOKEN / 16;
        int blocks = (waves * 32 + TB - 1) / TB;
        gemm_wmma_kernel<4, NHID><<<blocks, TB, 0, stream>>>(bufB, W2, bufA, NTOKEN);
    }
    // node_emb = dinv^2 * hw + b2   (overwrites bufB, h no longer needed)
    self_init_kernel<64><<<(int)(((long)NTOKEN * 64 + TB - 1) / TB), TB, 0, stream>>>(
        dinv, bufA, b2, bufB);
    edge_agg_kernel<64><<<(int)(((long)NEDGE * 32 + TB - 1) / TB), TB, 0, stream>>>(
        esrc, edst, dinv, bufA, bufB);

    // --- final gather:  out = node_emb[input] ----------------------------
    gather_kernel<<<(NROWS * 32 + TB - 1) / TB, TB, 0, stream>>>(input, bufB, out);
}